// CapsuleNet_3186865734190
// MI455X (gfx1250) — compile-verified
//
#include <hip/hip_runtime.h>

typedef float v2f __attribute__((ext_vector_type(2)));
typedef float v8f __attribute__((ext_vector_type(8)));

#define CAPS_NTHREADS 512
#define CAPS_NWAVE 16
#define CAPS_MB 16

// One workgroup = 16 batch rows, all O capsules, full dynamic routing fused.
// x-hat is never materialized: recomputed per (o,i) tile with V_WMMA_F32_16X16X4_F32.
// Persistent routing state (cumulative outputs P, softmax stats) lives in LDS.
template<int I, int Di, int O, int Do, bool LAST>
__global__ __launch_bounds__(CAPS_NTHREADS)
void caps_layer(const float* __restrict__ Xg,
                const float* __restrict__ Wg,
                float* __restrict__ Hout) {
  constexpr int NT = Do / 16;   // N tiles of 16 over Do
  constexpr int KT = Di / 4;    // K tiles of 4 over Di

  __shared__ float xs[CAPS_MB * I * Di];        // staged input tile  [b][i][di]
  __shared__ float P_lds[CAPS_MB * O * Do];     // cumulative outputs [b][o][do]
  __shared__ float s_lds[CAPS_MB * O * Do];     // routed sum         [b][o][do]
  __shared__ float mx_lds[CAPS_MB * I];         // softmax max  per (b,i)
  __shared__ float rse_lds[CAPS_MB * I];        // 1/sum(exp)   per (b,i)
  __shared__ float wscr[CAPS_NWAVE][16 * 16];   // per-wave D-tile bounce
  __shared__ float c_s[CAPS_NWAVE][16];         // per-wave routing coeff bcast

  const int tid  = (int)threadIdx.x;
  const int lane = tid & 31;
  const int wave = tid >> 5;
  const int row  = lane & 15;   // A row == B col == lane%16
  const int hi   = lane >> 4;   // lane half selects K pair / M+8 rows
  const int bg0  = (int)blockIdx.x * CAPS_MB;

  const v8f vzero = {0.f,0.f,0.f,0.f,0.f,0.f,0.f,0.f};

  // stage x tile into LDS (contiguous copy, coalesced)
  for (int t0 = tid; t0 < CAPS_MB * I * Di; t0 += CAPS_NTHREADS)
    xs[t0] = Xg[(size_t)bg0 * I * Di + t0];
  __syncthreads();

  // x_hat[b, nt*16+n] tile for capsule pair (o,i):  X_i[16xDi] @ W[o,i]^T[DixDo]
  auto compute_xh = [&](int o, int i, v8f* xh) {
    const float* xrow  = &xs[(row * I + i) * Di];
    const float* wbase = &Wg[(size_t)((o * I + i) * Do) * Di];
#pragma unroll
    for (int nt = 0; nt < NT; ++nt) {
      v8f acc = vzero;
#pragma unroll
      for (int kt = 0; kt < KT; ++kt) {
        const int k0 = kt * 4 + hi * 2;               // ISA f32 A/B lane layout
        v2f a; a.x = xrow[k0]; a.y = xrow[k0 + 1];
        const float* wp = wbase + (nt * 16 + row) * Di + k0;
        v2f b; b.x = wp[0]; b.y = wp[1];
        acc = __builtin_amdgcn_wmma_f32_16x16x4_f32(
            false, a, false, b, (short)0, acc, false, false);
      }
      xh[nt] = acc;
    }
  };

  // routing logit  bb[b] = P[b,o,:] . xh[b,:]   (valid in lanes 0..15, b=row)
  auto tile_bb = [&](const v8f* xh, int o) -> float {
    float bb = 0.f;
#pragma unroll
    for (int nt = 0; nt < NT; ++nt) {
#pragma unroll
      for (int v = 0; v < 8; ++v)
        wscr[wave][(v + 8 * hi) * 16 + row] = xh[nt][v];  // D row=v+8*hi, col=row
      __threadfence_block();
      if (lane < 16) {
        const float* prow = &P_lds[(row * O + o) * Do + nt * 16];
        const float* srow = &wscr[wave][row * 16];
#pragma unroll
        for (int c2 = 0; c2 < 16; ++c2) bb += prow[c2] * srow[c2];
      }
      __threadfence_block();
    }
    return bb;
  };

  for (int t = 0; t < 3; ++t) {
    // ---- phase A: s[b,o,:] = sum_i c[b,o,i] * x_hat[b,o,i,:]  (waves own o) ----
    for (int o = wave; o < O; o += CAPS_NWAVE) {
      v8f sacc[NT];
#pragma unroll
      for (int nt = 0; nt < NT; ++nt) sacc[nt] = vzero;
      for (int i = 0; i < I; ++i) {
        v8f xh[NT];
        compute_xh(o, i, xh);
        float crow[8];
        if (t == 0) {
#pragma unroll
          for (int v = 0; v < 8; ++v) crow[v] = 1.0f / (float)O;  // softmax(0)
        } else {
          const float bb = tile_bb(xh, o);
          if (lane < 16)
            c_s[wave][row] = __expf(bb - mx_lds[row * I + i]) * rse_lds[row * I + i];
          __threadfence_block();
#pragma unroll
          for (int v = 0; v < 8; ++v) crow[v] = c_s[wave][v + 8 * hi];
          __threadfence_block();
        }
#pragma unroll
        for (int nt = 0; nt < NT; ++nt)
#pragma unroll
          for (int v = 0; v < 8; ++v)
            sacc[nt][v] += crow[v] * xh[nt][v];
      }
#pragma unroll
      for (int nt = 0; nt < NT; ++nt)
#pragma unroll
        for (int v = 0; v < 8; ++v)
          s_lds[((v + 8 * hi) * O + o) * Do + nt * 16 + row] = sacc[nt][v];
    }
    __syncthreads();

    // ---- phase B: squash + cumulative-output update / final emit ----
    for (int idx = tid; idx < CAPS_MB * O; idx += CAPS_NTHREADS) {
      const int b = idx / O;
      const int o = idx - b * O;
      const float* sp = &s_lds[(b * O + o) * Do];
      float n2 = 0.f;
#pragma unroll 4
      for (int d = 0; d < Do; ++d) n2 += sp[d] * sp[d];
      const float nrm   = sqrtf(n2);
      const float scale = n2 / ((1.f + n2) * (nrm + 1e-8f));
      if (t < 2) {
        float* pp = &P_lds[(b * O + o) * Do];
#pragma unroll 4
        for (int d = 0; d < Do; ++d) {
          const float ov = scale * sp[d];
          pp[d] = (t == 0) ? ov : (pp[d] + ov);   // P_t = sum of outputs so far
        }
      } else if (LAST) {
        Hout[(size_t)(bg0 + b) * O + o] = scale * nrm;   // ||squash(s)||
      } else {
        float* hp = &Hout[((size_t)(bg0 + b) * O + o) * Do];
#pragma unroll 4
        for (int d = 0; d < Do; ++d) hp[d] = scale * sp[d];
      }
    }
    __syncthreads();

    // ---- phase C: online softmax stats over o for next iter (waves own i) ----
    if (t < 2) {
      for (int i = wave; i < I; i += CAPS_NWAVE) {
        float mx = -3.0e38f;
        float se = 0.f;
        for (int o = 0; o < O; ++o) {
          v8f xh[NT];
          compute_xh(o, i, xh);
          const float bb = tile_bb(xh, o);
          if (lane < 16) {
            const float nm = fmaxf(mx, bb);
            se = se * __expf(mx - nm) + __expf(bb - nm);
            mx = nm;
          }
        }
        if (lane < 16) {
          mx_lds[row * I + i]  = mx;
          rse_lds[row * I + i] = 1.f / se;
        }
      }
      __syncthreads();
    }
  }
}

extern "C" void kernel_launch(void* const* d_in, const int* in_sizes, int n_in,
                              void* d_out, int out_size, void* d_ws, size_t ws_size,
                              hipStream_t stream) {
  (void)in_sizes; (void)n_in; (void)out_size; (void)ws_size;
  const float* x  = (const float*)d_in[0];   // [1024,1024] == [1024,128,8]
  const float* W0 = (const float*)d_in[1];   // [64,128,16,8]
  const float* W1 = (const float*)d_in[2];   // [32,64,32,16]
  const float* W2 = (const float*)d_in[3];   // [10,32,64,32]
  float* h1  = (float*)d_ws;                 // [1024,64,16]  (4 MB)
  float* h2  = h1 + 1024 * 64 * 16;          // [1024,32,32]  (4 MB)
  float* out = (float*)d_out;                // [1024,10]

  dim3 grid(1024 / CAPS_MB), block(CAPS_NTHREADS);
  caps_layer<128,  8, 64, 16, false><<<grid, block, 0, stream>>>(x,  W0, h1);
  caps_layer< 64, 16, 32, 32, false><<<grid, block, 0, stream>>>(h1, W1, h2);
  caps_layer< 32, 32, 10, 64, true ><<<grid, block, 0, stream>>>(h2, W2, out);
}